// MarginLoss_74758200754289
// MI455X (gfx1250) — compile-verified
//
#include <hip/hip_runtime.h>
#include <hip/hip_bf16.h>
#include <stdint.h>

typedef float v2f __attribute__((ext_vector_type(2)));
typedef float v8f __attribute__((ext_vector_type(8)));

#define C_DIM 512
#define T_DIM 2048
#define V_DIM 4096
#define B_DIM 8
#define N_ROWS (B_DIM * T_DIM)            // 16384
#define ROWS_PER_BLOCK 32
#define NBLOCKS (N_ROWS / ROWS_PER_BLOCK) // 512
#define LDA 516                           // padded LDS row stride (floats): 516 % 64 == 4 -> conflict-free
#define MARGIN_F 0.5f
#define FLT_BIG 3.0e38f

// ---------------------------------------------------------------------------
// Kernel 1: codebook squared norms -> d_ws   (deterministic tree reduce)
// ---------------------------------------------------------------------------
__global__ __launch_bounds__(128)
void vq_bnorm_kernel(const float* __restrict__ cb, float* __restrict__ bnorm) {
    __shared__ float red[128];
    const int v = blockIdx.x;
    const float* row = cb + (size_t)v * C_DIM;
    float s = 0.0f;
    for (int c = threadIdx.x; c < C_DIM; c += 128) {
        float x = row[c];
        s += x * x;
    }
    red[threadIdx.x] = s;
    __syncthreads();
    for (int off = 64; off > 0; off >>= 1) {
        if (threadIdx.x < off) red[threadIdx.x] += red[threadIdx.x + off];
        __syncthreads();
    }
    if (threadIdx.x == 0) bnorm[v] = red[0];
}

// ---------------------------------------------------------------------------
// Kernel 2: fused distance-GEMM (fp32 WMMA, 2x2 register blocking) +
// masked row-min. One block = 32 student rows (two 16-row WMMA tiles in LDS).
// 8 waves; each wave sweeps 16 pairs of 16-col codebook tiles; per k-step
// 2 LDS A-frags + 2 global B-frags feed 4 independent WMMA chains.
// Correct-code distance computed directly in the prologue (1 dot/row).
// ---------------------------------------------------------------------------
__global__ __launch_bounds__(256)
void vq_margin_main_kernel(const float* __restrict__ emb,
                           const int*   __restrict__ codes,
                           const float* __restrict__ cb,
                           const float* __restrict__ bnorm,
                           float*       __restrict__ partial) {
    __shared__ float Atile[ROWS_PER_BLOCK * LDA];   // 66 KB: A[m][c], padded
    __shared__ float norm_part[ROWS_PER_BLOCK * 8];
    __shared__ float dot_part[ROWS_PER_BLOCK * 8];
    __shared__ float anorm[ROWS_PER_BLOCK];
    __shared__ float correctd2[ROWS_PER_BLOCK];
    __shared__ int   wrongmin_i[ROWS_PER_BLOCK];    // float-as-int (vals >= 0)
    __shared__ int   codeL[ROWS_PER_BLOCK];
    __shared__ float rowloss[ROWS_PER_BLOCK];

    const int tid = threadIdx.x;
    const int n0  = blockIdx.x * ROWS_PER_BLOCK;    // first flat row (= b*T + t0)
    const int b   = n0 / T_DIM;
    const int t0  = n0 % T_DIM;                     // 32 | T -> whole block in one b
    const float* embB = emb + (size_t)b * C_DIM * T_DIM + t0;

    // ---- stage A via async Global->LDS: A[m][c] = student_emb[b][c][t0+m] ----
    {
        const int m  = tid & 31;
        const int c0 = tid >> 5;                    // 0..7
        #pragma unroll
        for (int j = 0; j < C_DIM / 8; ++j) {
            const int c = c0 + 8 * j;
            const float* gaddr = embB + (size_t)c * T_DIM + m;
            // low 32 bits of the flat shared-aperture address == LDS byte offset
            unsigned lds_off = (unsigned)(uintptr_t)(&Atile[m * LDA + c]);
            asm volatile("global_load_async_to_lds_b32 %0, %1, off"
                         :: "v"(lds_off), "v"(gaddr)
                         : "memory");
        }
        asm volatile("s_wait_asynccnt 0x0" ::: "memory");
    }
    if (tid < ROWS_PER_BLOCK) {
        codeL[tid]      = codes[n0 + tid];
        wrongmin_i[tid] = 0x7F7FFFFF;               // +FLT_MAX bits
    }
    __syncthreads();

    // ---- prologue: per-row ||a||^2 and correct-code dot (deterministic) ----
    {
        const int r = tid >> 3, li = tid & 7;       // r: 0..31, li: 0..7
        const float* cbrow = cb + (size_t)codeL[r] * C_DIM;
        float sn = 0.0f, sd = 0.0f;
        #pragma unroll
        for (int j = 0; j < C_DIM / 8; ++j) {
            const int c = li + 8 * j;
            const float a = Atile[r * LDA + c];
            sn += a * a;
            sd += a * cbrow[c];
        }
        norm_part[r * 8 + li] = sn;
        dot_part[r * 8 + li]  = sd;
    }
    __syncthreads();
    if (tid < ROWS_PER_BLOCK) {
        float sn = 0.0f, sd = 0.0f;
        #pragma unroll
        for (int j = 0; j < 8; ++j) {
            sn += norm_part[tid * 8 + j];
            sd += dot_part[tid * 8 + j];
        }
        anorm[tid] = sn;
        correctd2[tid] = fmaxf(sn + bnorm[codeL[tid]] - 2.0f * sd, 0.0f);
    }
    __syncthreads();

    // ---- WMMA sweep over codebook, 2x2 blocked ----
    const int wave  = tid >> 5;                     // 0..7 (wave32)
    const int lane  = tid & 31;
    const int lhalf = lane >> 4;                    // 0 or 1
    const int lmod  = lane & 15;

    // A-operand ISA layout (16x4 f32): lanes 0-15 hold K={0,1}, lanes 16-31 K={2,3}
    const float* Abase0 = &Atile[lmod * LDA + 2 * lhalf];
    const float* Abase1 = Abase0 + 16 * LDA;

    // hoist per-row constants for the 16 rows this lane's accumulators cover
    float an_r[16];
    int   cd_r[16];
    #pragma unroll
    for (int r = 0; r < 8; ++r) {
        an_r[r]     = anorm[r + 8 * lhalf];
        cd_r[r]     = codeL[r + 8 * lhalf];
        an_r[8 + r] = anorm[16 + r + 8 * lhalf];
        cd_r[8 + r] = codeL[16 + r + 8 * lhalf];
    }

    float lw[16];
    #pragma unroll
    for (int r = 0; r < 16; ++r) lw[r] = FLT_BIG;

    for (int it = 0; it < 16; ++it) {
        const int p     = wave + 8 * it;            // 0..127 column-pair index
        const int ncol0 = p * 32 + lmod;
        const int ncol1 = ncol0 + 16;
        // B-operand (4x16): lane n holds B[K={0,1}][n] / B[K={2,3}][n]
        const float* Bbase0 = cb + (size_t)ncol0 * C_DIM + 2 * lhalf;
        const float* Bbase1 = Bbase0 + 16 * C_DIM;  // +32 KB: same addr VGPR + imm offset

        v8f a00 = {}, a01 = {}, a10 = {}, a11 = {};
        #pragma unroll 4
        for (int kk = 0; kk < 128; ++kk) {
            v2f av0 = *(const v2f*)(Abase0 + 4 * kk);
            v2f av1 = *(const v2f*)(Abase1 + 4 * kk);
            v2f bv0 = *(const v2f*)(Bbase0 + 4 * kk);
            v2f bv1 = *(const v2f*)(Bbase1 + 4 * kk);
            a00 = __builtin_amdgcn_wmma_f32_16x16x4_f32(false, av0, false, bv0, (short)0, a00, false, false);
            a01 = __builtin_amdgcn_wmma_f32_16x16x4_f32(false, av0, false, bv1, (short)0, a01, false, false);
            a10 = __builtin_amdgcn_wmma_f32_16x16x4_f32(false, av1, false, bv0, (short)0, a10, false, false);
            a11 = __builtin_amdgcn_wmma_f32_16x16x4_f32(false, av1, false, bv1, (short)0, a11, false, false);
        }

        const float bn0 = bnorm[ncol0];
        const float bn1 = bnorm[ncol1];
        // branchless epilogue: masked running-min only
        #pragma unroll
        for (int r = 0; r < 8; ++r) {
            float d2;
            d2 = fmaxf(an_r[r] + bn0 - 2.0f * a00[r], 0.0f);
            lw[r] = fminf(lw[r], (ncol0 == cd_r[r]) ? FLT_BIG : d2);
            d2 = fmaxf(an_r[r] + bn1 - 2.0f * a01[r], 0.0f);
            lw[r] = fminf(lw[r], (ncol1 == cd_r[r]) ? FLT_BIG : d2);
            d2 = fmaxf(an_r[8 + r] + bn0 - 2.0f * a10[r], 0.0f);
            lw[8 + r] = fminf(lw[8 + r], (ncol0 == cd_r[8 + r]) ? FLT_BIG : d2);
            d2 = fmaxf(an_r[8 + r] + bn1 - 2.0f * a11[r], 0.0f);
            lw[8 + r] = fminf(lw[8 + r], (ncol1 == cd_r[8 + r]) ? FLT_BIG : d2);
        }
    }

    // ---- combine: order-independent min on non-negative float bits ----
    #pragma unroll
    for (int r = 0; r < 8; ++r) {
        atomicMin(&wrongmin_i[r + 8 * lhalf],      __float_as_int(lw[r]));
        atomicMin(&wrongmin_i[16 + r + 8 * lhalf], __float_as_int(lw[8 + r]));
    }
    __syncthreads();

    if (tid < ROWS_PER_BLOCK) {
        const float cd = sqrtf(correctd2[tid]);
        const float wd = sqrtf(__int_as_float(wrongmin_i[tid]));
        rowloss[tid] = fmaxf(cd - wd + MARGIN_F, 0.0f);
    }
    __syncthreads();
    if (tid == 0) {
        float s = 0.0f;
        #pragma unroll
        for (int j = 0; j < ROWS_PER_BLOCK; ++j) s += rowloss[j];
        partial[blockIdx.x] = s;
    }
}

// ---------------------------------------------------------------------------
// Kernel 3: deterministic final reduction -> mean loss
// ---------------------------------------------------------------------------
__global__ __launch_bounds__(256)
void vq_reduce_kernel(const float* __restrict__ partial, float* __restrict__ out) {
    __shared__ float red[256];
    float s = 0.0f;
    for (int i = threadIdx.x; i < NBLOCKS; i += 256) s += partial[i];
    red[threadIdx.x] = s;
    __syncthreads();
    for (int off = 128; off > 0; off >>= 1) {
        if (threadIdx.x < off) red[threadIdx.x] += red[threadIdx.x + off];
        __syncthreads();
    }
    if (threadIdx.x == 0) out[0] = red[0] / (float)N_ROWS;
}

// ---------------------------------------------------------------------------
extern "C" void kernel_launch(void* const* d_in, const int* in_sizes, int n_in,
                              void* d_out, int out_size, void* d_ws, size_t ws_size,
                              hipStream_t stream) {
    const float* emb   = (const float*)d_in[0];   // (B, C, T) fp32
    const int*   codes = (const int*)  d_in[1];   // (B, T) int
    const float* cb    = (const float*)d_in[2];   // (V, C) fp32

    float* bnorm   = (float*)d_ws;                // V floats
    float* partial = bnorm + V_DIM;               // NBLOCKS floats
    float* out     = (float*)d_out;

    hipLaunchKernelGGL(vq_bnorm_kernel, dim3(V_DIM), dim3(128), 0, stream, cb, bnorm);
    hipLaunchKernelGGL(vq_margin_main_kernel, dim3(NBLOCKS), dim3(256), 0, stream,
                       emb, codes, cb, bnorm, partial);
    hipLaunchKernelGGL(vq_reduce_kernel, dim3(1), dim3(256), 0, stream, partial, out);
}